// _split_decision_19670950215707
// MI455X (gfx1250) — compile-verified
//
#include <hip/hip_runtime.h>

#define N_SAMPLES   1000000
#define N_FEAT      100
#define MAX_BIN     256
#define HIST_ELEMS  (2 * N_FEAT * MAX_BIN)   /* 51200 f32 = 200 KB */
#define H_OFF       (N_FEAT * MAX_BIN)       /* 25600: hessian half  */
#define MAX_PARTIAL 256

typedef float v2f __attribute__((ext_vector_type(2)));
typedef float v8f __attribute__((ext_vector_type(8)));

// -------------------------------------------------------------------------
// Kernel 1: per-WG privatized histograms in 200 KB of LDS (CDNA5: 320 KB/WG).
// LDS layout: [value(2)][bin(256)][feature(100)] so consecutive lanes
// (consecutive features) hit consecutive LDS banks on the ds_add_f32 path.
// X rows (400 B) are read as 25 coalesced int4 loads per wave.
// -------------------------------------------------------------------------
__global__ void __launch_bounds__(1024)
hist_kernel(const int* __restrict__ X, const float* __restrict__ grad,
            const float* __restrict__ hess, float* __restrict__ partials,
            int samples_per_wg)
{
    extern __shared__ float smem[];                 // HIST_ELEMS floats (dynamic)
    const int tid = threadIdx.x;

    for (int e = tid; e < HIST_ELEMS; e += blockDim.x) smem[e] = 0.0f;
    __syncthreads();

    const int lane        = tid & 31;
    const int waveInWg    = tid >> 5;               // 0..31 (wave32)
    const int wavesPerWg  = blockDim.x >> 5;

    long start = (long)blockIdx.x * samples_per_wg;
    long end   = start + samples_per_wg;
    if (end > N_SAMPLES) end = N_SAMPLES;

    for (long i = start + waveInWg; i < end; i += wavesPerWg) {
        const float gv = grad[i];                   // same addr across wave -> 1 req
        const float hv = hess[i];
        if (lane < 25) {                            // 25 lanes * int4 = one 400B row
            const int4 b = *(reinterpret_cast<const int4*>(X + (size_t)i * N_FEAT) + lane);
            const int f0 = lane * 4;
            atomicAdd(&smem[b.x * N_FEAT + (f0 + 0)], gv);
            atomicAdd(&smem[b.y * N_FEAT + (f0 + 1)], gv);
            atomicAdd(&smem[b.z * N_FEAT + (f0 + 2)], gv);
            atomicAdd(&smem[b.w * N_FEAT + (f0 + 3)], gv);
            atomicAdd(&smem[H_OFF + b.x * N_FEAT + (f0 + 0)], hv);
            atomicAdd(&smem[H_OFF + b.y * N_FEAT + (f0 + 1)], hv);
            atomicAdd(&smem[H_OFF + b.z * N_FEAT + (f0 + 2)], hv);
            atomicAdd(&smem[H_OFF + b.w * N_FEAT + (f0 + 3)], hv);
        }
    }
    __syncthreads();

    float* dst = partials + (size_t)blockIdx.x * HIST_ELEMS;
    for (int e = tid; e < HIST_ELEMS; e += blockDim.x) dst[e] = smem[e];
}

// -------------------------------------------------------------------------
// Kernel 2: deterministic reduction of partials -> hist (fully coalesced).
// -------------------------------------------------------------------------
__global__ void reduce_kernel(const float* __restrict__ partials,
                              float* __restrict__ hist, int nwg)
{
    const int e = blockIdx.x * blockDim.x + threadIdx.x;
    if (e >= HIST_ELEMS) return;
    float acc = 0.0f;
    for (int w = 0; w < nwg; ++w)
        acc += partials[(size_t)w * HIST_ELEMS + e];
    hist[e] = acc;
}

// -------------------------------------------------------------------------
// Kernel 3: cumsum over bins as out = hist * T (T lower-unit-triangular ones)
// using V_WMMA_F32_16X16X4_F32 in full f32 precision.
//   rows  R = v*100 + f  (200 real rows, padded to 13 tiles of 16)
//   A tile: 16 rows x 4 K-bins per step; B synthesized in registers:
//     all-ones for K-blocks strictly left of the column tile,
//     (k_global <= n_global) inside the diagonal block, zero (skipped) after.
// VGPR layouts per ISA 7.12.2:
//   A : m = lane&15, K = {0,1} (lanes 0-15) / {2,3} (lanes 16-31)
//   B : n = lane&15, same K split
//   C/D elem r : M = r + (lane<16 ? 0 : 8), N = lane&15
// -------------------------------------------------------------------------
#define TILE_PITCH 257   /* 257 % 64 == 1 -> conflict-free column reads */

__global__ void cumsum_wmma_kernel(const float* __restrict__ hist,
                                   float* __restrict__ out)
{
    __shared__ float tile[16 * TILE_PITCH];         // ~16.4 KB
    const int rowBase = blockIdx.x * 16;

    // Stage tile[r][bin] = hist[v*25600 + bin*100 + f]; pad rows >= 200 with 0.
    for (int idx = threadIdx.x; idx < 16 * 256; idx += blockDim.x) {
        const int r = idx >> 8, bin = idx & 255;
        const int R = rowBase + r;
        float val = 0.0f;
        if (R < 2 * N_FEAT) {
            const int v = R / N_FEAT, f = R % N_FEAT;
            val = hist[v * H_OFF + bin * N_FEAT + f];
        }
        tile[r * TILE_PITCH + bin] = val;
    }
    __syncthreads();

    if (threadIdx.x < 32) {                         // one full wave, EXEC all-1s
        const int lane  = threadIdx.x;
        const int n     = lane & 15;                // N (and M for A) index
        const int kOff  = (lane < 16) ? 0 : 2;      // K pair held by this lane
        const int mBase = (lane < 16) ? 0 : 8;      // C/D row split

        for (int j = 0; j < 16; ++j) {              // 16-bin column tiles
            v8f c = {0.f, 0.f, 0.f, 0.f, 0.f, 0.f, 0.f, 0.f};
            const int kBlocks = 4 * j + 4;          // K-blocks with nonzero T
            for (int k4 = 0; k4 < kBlocks; ++k4) {
                v2f a;
                a.x = tile[n * TILE_PITCH + 4 * k4 + kOff];
                a.y = tile[n * TILE_PITCH + 4 * k4 + kOff + 1];
                v2f b;
                if (k4 < 4 * j) {                   // fully below diagonal
                    b.x = 1.0f; b.y = 1.0f;
                } else {                            // diagonal block t = 0..3
                    const int t = k4 - 4 * j;
                    b.x = (4 * t + kOff     <= n) ? 1.0f : 0.0f;
                    b.y = (4 * t + kOff + 1 <= n) ? 1.0f : 0.0f;
                }
                c = __builtin_amdgcn_wmma_f32_16x16x4_f32(
                        false, a, false, b, (short)0, c, false, false);
            }
#pragma unroll
            for (int r = 0; r < 8; ++r) {
                const int R = rowBase + mBase + r;
                if (R < 2 * N_FEAT) {
                    const int v = R / N_FEAT, f = R % N_FEAT;
                    out[v * H_OFF + f * MAX_BIN + 16 * j + n] = c[r];
                }
            }
        }
    }
}

// -------------------------------------------------------------------------
extern "C" void kernel_launch(void* const* d_in, const int* in_sizes, int n_in,
                              void* d_out, int out_size, void* d_ws, size_t ws_size,
                              hipStream_t stream)
{
    const int*   X    = (const int*)d_in[0];
    const float* grad = (const float*)d_in[1];
    const float* hess = (const float*)d_in[2];
    float*       out  = (float*)d_out;
    float*       ws   = (float*)d_ws;

    // ws layout: [0, HIST_ELEMS) reduced hist | [HIST_ELEMS, ...) nwg partials
    long slots = (long)(ws_size / sizeof(float)) / HIST_ELEMS - 1;
    int  nwg   = (int)(slots < 1 ? 1 : (slots > MAX_PARTIAL ? MAX_PARTIAL : slots));

    float* hist     = ws;
    float* partials = ws + HIST_ELEMS;
    const int samples_per_wg = (N_SAMPLES + nwg - 1) / nwg;

    hist_kernel<<<nwg, 1024, HIST_ELEMS * sizeof(float), stream>>>(
        X, grad, hess, partials, samples_per_wg);

    reduce_kernel<<<(HIST_ELEMS + 255) / 256, 256, 0, stream>>>(partials, hist, nwg);

    cumsum_wmma_kernel<<<(2 * N_FEAT + 15) / 16, 128, 0, stream>>>(hist, out);
}